// DeepMFMVDR_59880434040859
// MI455X (gfx1250) — compile-verified
//
#include <hip/hip_runtime.h>
#include <stdint.h>

// DeepMFMVDR on MI455X (gfx1250).
// One thread per (b,f,t); 520k independent 5x5 complex MVDR problems.
// Roofline: ~115 MB I/O @ 23.3 TB/s => ~5us; ~1.9 GFLOP of VALU => balanced.
// WMMA is intentionally NOT used: per-element 5x5 matmuls cannot share WMMA's
// jointly-summed K dim without block-diagonal packing (<=18% useful MACs) plus
// cross-lane relayout costs that exceed the FMAs saved. The CDNA5-specific
// paths used: async global->LDS staging (ASYNCcnt) of the overlapping 5-frame
// y window, and NT cache policy on single-use streaming data.

#define TT 1000                 // time frames (fixed by harness setup_inputs)
#define LF 5                    // filter length

typedef float v4f __attribute__((ext_vector_type(4)));
typedef v4f v4f_u __attribute__((aligned(4)));     // unaligned-capable float4
typedef float v2f __attribute__((ext_vector_type(2)));

struct Cx { float r, i; };

__device__ __forceinline__ Cx cmul(Cx a, Cx b) {
  return { fmaf(a.r, b.r, -a.i * b.i), fmaf(a.r, b.i, a.i * b.r) };
}
__device__ __forceinline__ Cx crecip(Cx d) {
  float s = 1.0f / fmaf(d.r, d.r, d.i * d.i);
  return { d.r * s, -d.i * s };
}

// 25 floats, single-use stream: non-temporal b128 packets (+1 scalar).
__device__ __forceinline__ void load25_nt(const float* p, float* v) {
  const v4f_u* p4 = (const v4f_u*)p;
#pragma unroll
  for (int q = 0; q < 6; q++) {
    v4f b = __builtin_nontemporal_load(&p4[q]);
    v[4 * q + 0] = b.x; v[4 * q + 1] = b.y;
    v[4 * q + 2] = b.z; v[4 * q + 3] = b.w;
  }
  v[24] = __builtin_nontemporal_load(p + 24);
}

// vec(25) -> full 5x5 complex Hermitian (reference _vec_to_hermitian layout)
__device__ __forceinline__ void expandH(const float* v, float ar[5][5], float ai[5][5]) {
  const int ru0[5] = {0, 5, 9, 12, 14};   // triu(incl diag) row starts
  const int ru1[4] = {0, 4, 7, 9};        // strict-triu row starts
#pragma unroll
  for (int i = 0; i < 5; i++) { ar[i][i] = v[ru0[i]]; ai[i][i] = 0.f; }
#pragma unroll
  for (int i = 0; i < 4; i++) {
#pragma unroll
    for (int j = i + 1; j < 5; j++) {
      float re = v[ru0[i] + (j - i)];
      float im = v[15 + ru1[i] + (j - i - 1)];
      ar[i][j] = re;  ai[i][j] = im;
      ar[j][i] = re;  ai[j][i] = -im;
    }
  }
}

// CDNA5 async global->LDS copy of one complex frame (8 bytes, ASYNCcnt).
__device__ __forceinline__ void stage_y(float2* sY, int e, uint64_t gbase, int t0) {
  int tg = t0 - 4 + e;                       // global time index of element e
  if (tg < 0) {                              // left zero-padding (t<0)
    sY[e].x = 0.f; sY[e].y = 0.f;
  } else if (tg < TT) {
    uint32_t lds = (uint32_t)(uintptr_t)(&sY[e]);   // flat low 32 bits == LDS byte addr
    int off = tg * 8;                                // byte offset into this (b,f) row
    asm volatile("global_load_async_to_lds_b64 %0, %1, %2"
                 :
                 : "v"(lds), "v"(off), "s"(gbase)
                 : "memory");
  }
}

__global__ __launch_bounds__(256) void
DeepMFMVDR_kernel(const float* __restrict__ noisy,
                  const float* __restrict__ phin_vec,
                  const float* __restrict__ phiy_vec,
                  const float* __restrict__ xi_raw,
                  float* __restrict__ out) {
  constexpr float EPSF = 1.1920929e-7f;          // float32 eps
  constexpr float REGC = 0.001f;
  constexpr float GMIN = 0.14125375446227544f;   // 10^(-17/20)
  constexpr float KG   = 10.0f;

  __shared__ float2 sY[256 + LF - 1];            // y[t0-4 .. t0+255]

  const int tid = threadIdx.x;
  const int bf  = blockIdx.x;                    // (b*F + f)
  const int t0  = blockIdx.y * 256;
  const int t   = t0 + tid;
  // Clamped element index: out-of-range lanes redundantly recompute t=TT-1
  // (free in lockstep) so all loads/compute run unconditionally under full
  // EXEC -- no entry branch, no else-path register zero-fill.
  const int tc = (t < TT) ? t : (TT - 1);
  const size_t elem = (size_t)bf * TT + tc;

  // ---- early-issue single-use streaming loads (overlap with barrier) ----
  float vy[25];
  load25_nt(phiy_vec + elem * 25, vy);
  float xr = __builtin_nontemporal_load(xi_raw + elem);

  // ---- async stage of the y window for this block ----
  {
    const uint64_t ybase = (uint64_t)(uintptr_t)(noisy + (size_t)bf * TT * 2);
    stage_y(sY, tid, ybase, t0);
    if (tid < LF - 1) stage_y(sY, 256 + tid, ybase, t0);
  }
  asm volatile("s_wait_asynccnt 0x0" ::: "memory");
  __syncthreads();

  // ---- a-priori SNR: softplus ----
  float xs = (xr > 20.f) ? xr : log1pf(__expf(xr));
  float c2 = 1.f / (xs + EPSF);
  float c1 = (1.f + xs) * c2;

  // ---- gammay: only last column + trace of Phiy needed ----
  Cx gy[5];
  {
    float ayr[5][5], ayi[5][5];
    expandH(vy, ayr, ayi);

    float trY = 0.f;                           // trace(Ay Ay^H) = ||Ay||_F^2
#pragma unroll
    for (int i = 0; i < 5; i++)
#pragma unroll
      for (int k = 0; k < 5; k++)
        trY = fmaf(ayr[i][k], ayr[i][k], fmaf(ayi[i][k], ayi[i][k], trY));

    Cx col[5];                                 // Phiy[:,4] = Ay * Ay[:,4] (Ay Hermitian)
#pragma unroll
    for (int i = 0; i < 5; i++) {
      float sr = 0.f, si = 0.f;
#pragma unroll
      for (int k = 0; k < 5; k++) {
        sr = fmaf(ayr[i][k], ayr[k][4], fmaf(-ayi[i][k], ayi[k][4], sr));
        si = fmaf(ayr[i][k], ayi[k][4], fmaf( ayi[i][k], ayr[k][4], si));
      }
      col[i].r = sr; col[i].i = si;
    }
    float sclY = fmaf(REGC * 0.2f, trY, EPSF); // reg*tr/L + eps
    col[4].r += sclY;                          // tik_reg touches diagonal only
    Cx rY = crecip({col[4].r + EPSF, col[4].i});
#pragma unroll
    for (int i = 0; i < 5; i++) gy[i] = cmul(col[i], rY);
  }

  // ---- Phin full (Hermitian product, upper triangle + mirror) ----
  float Pr[5][5], Pi[5][5];
  {
    float vn[25];
    load25_nt(phin_vec + elem * 25, vn);
    float anr[5][5], ani[5][5];
    expandH(vn, anr, ani);

    float tr = 0.f;
#pragma unroll
    for (int i = 0; i < 5; i++) {
#pragma unroll
      for (int j = i; j < 5; j++) {
        float sr = 0.f, si = 0.f;
#pragma unroll
        for (int k = 0; k < 5; k++) {
          sr = fmaf(anr[i][k], anr[k][j], fmaf(-ani[i][k], ani[k][j], sr));
          si = fmaf(anr[i][k], ani[k][j], fmaf( ani[i][k], anr[k][j], si));
        }
        Pr[i][j] = sr; Pi[i][j] = si;
        if (j > i) { Pr[j][i] = sr; Pi[j][i] = -si; } else { tr += sr; }
      }
    }
    float scl = fmaf(REGC * 0.2f, tr, EPSF);
#pragma unroll
    for (int i = 0; i < 5; i++) Pr[i][i] += scl;
  }

  // ---- gamman and gammax ----
  Cx gx[5];
  {
    Cx rN = crecip({Pr[4][4] + EPSF, Pi[4][4]});
#pragma unroll
    for (int i = 0; i < 5; i++) {
      Cx gn = cmul({Pr[i][4], Pi[i][4]}, rN);
      gx[i].r = fmaf(c1, gy[i].r, -c2 * gn.r);
      gx[i].i = fmaf(c1, gy[i].i, -c2 * gn.i);
    }
  }

  // ---- solve Phin * x = gx (unpivoted LU; Phin is HPD) ----
  Cx g[5];
#pragma unroll
  for (int i = 0; i < 5; i++) g[i] = gx[i];
#pragma unroll
  for (int k = 0; k < 5; k++) {
    Cx ip = crecip({Pr[k][k], Pi[k][k]});
#pragma unroll
    for (int i = k + 1; i < 5; i++) {
      Cx m = cmul({Pr[i][k], Pi[i][k]}, ip);
#pragma unroll
      for (int j = k + 1; j < 5; j++) {
        Cx tt = cmul(m, {Pr[k][j], Pi[k][j]});
        Pr[i][j] -= tt.r; Pi[i][j] -= tt.i;
      }
      Cx tg = cmul(m, g[k]);
      g[i].r -= tg.r; g[i].i -= tg.i;
    }
  }
  Cx x[5];
#pragma unroll
  for (int i = 4; i >= 0; i--) {
    Cx s = g[i];
#pragma unroll
    for (int j = i + 1; j < 5; j++) {
      Cx tt = cmul({Pr[i][j], Pi[i][j]}, x[j]);
      s.r -= tt.r; s.i -= tt.i;
    }
    x[i] = cmul(s, crecip({Pr[i][i], Pi[i][i]}));
  }

  // ---- MVDR weights: w = x / (gx^H x + eps) ----
  float dr = EPSF, di = 0.f;
#pragma unroll
  for (int i = 0; i < 5; i++) {
    dr = fmaf(gx[i].r, x[i].r, fmaf( gx[i].i, x[i].i, dr));
    di = fmaf(gx[i].r, x[i].i, fmaf(-gx[i].i, x[i].r, di));
  }
  Cx rd = crecip({dr, di});

  // ---- FIR over 5-frame window (from LDS) + minimum-gain blend ----
  float for_ = 0.f, foi = 0.f;
#pragma unroll
  for (int l = 0; l < 5; l++) {
    Cx w = cmul(x[l], rd);
    float2 yv = sY[tid + l];
    for_ = fmaf(w.r, yv.x, fmaf( w.i, yv.y, for_));   // conj(w)*y
    foi  = fmaf(w.r, yv.y, fmaf(-w.i, yv.x, foi));
  }
  float2 yc = sY[tid + 4];
  float fir = GMIN * yc.x, fii = GMIN * yc.y;
  float mo = sqrtf(fmaf(for_, for_, foi * foi));
  float mi = sqrtf(fmaf(fir, fir, fii * fii));
  float a  = 1.0f / (1.0f + __expf(-KG * (mo - mi)));
  float er = fmaf(a, for_ - fir, fir);
  float ei = fmaf(a, foi - fii, fii);

  if (t < TT) {                                  // only the store is guarded
    v2f ev; ev.x = er; ev.y = ei;                // 8B-aligned NT store
    __builtin_nontemporal_store(ev, (v2f*)(out + elem * 2));
  }
}

extern "C" void kernel_launch(void* const* d_in, const int* in_sizes, int n_in,
                              void* d_out, int out_size, void* d_ws, size_t ws_size,
                              hipStream_t stream) {
  (void)n_in; (void)out_size; (void)d_ws; (void)ws_size;
  const float* noisy    = (const float*)d_in[0];
  const float* phin_vec = (const float*)d_in[1];
  const float* phiy_vec = (const float*)d_in[2];
  const float* xi_raw   = (const float*)d_in[3];
  float* out = (float*)d_out;

  const int BF = in_sizes[3] / TT;                 // B*F = 520
  dim3 grid(BF, (TT + 255) / 256);                 // (520, 4)
  DeepMFMVDR_kernel<<<grid, 256, 0, stream>>>(noisy, phin_vec, phiy_vec, xi_raw, out);
}